// CenterHead_11639361372825
// MI455X (gfx1250) — compile-verified
//
#include <hip/hip_runtime.h>
#include <stdint.h>

#define BATCH 16
#define CHN   3
#define HH    512
#define WW    512
#define HW    (HH * WW)

#define NBINS     8192
#define BIN_SHIFT 19
#define CAP       4096

#define TX 64
#define TY 16
#define HX (TX + 2)   // 66
#define HY (TY + 2)   // 18
#define LSTRIDE 68    // padded LDS row stride in floats

// Monotonic order-preserving key for float32 (masked-out pixels use key 0,
// strictly below ord_key of any finite float).
__device__ __forceinline__ uint32_t ord_key(float x) {
  uint32_t b = __float_as_uint(x);
  return (b & 0x80000000u) ? ~b : (b | 0x80000000u);
}

// Load a (TX+2)x(TY+2) halo tile of one plane into LDS.
// In-range elements use the CDNA5 async global->LDS DMA path (ASYNCcnt),
// out-of-range halo gets -inf (reduce_window 'SAME' identity).
__device__ __forceinline__ void load_tile(const float* __restrict__ plane,
                                          int x0, int y0, float* tile) {
  const int tid = threadIdx.x;
  for (int e = tid; e < HX * HY; e += 256) {
    int hy = e / HX;
    int hx = e - hy * HX;
    int gx = x0 + hx - 1;
    int gy = y0 + hy - 1;
    int lofs = hy * LSTRIDE + hx;
    if ((unsigned)gx < (unsigned)WW && (unsigned)gy < (unsigned)HH) {
      uint32_t lds_byte = (uint32_t)(uintptr_t)(tile + lofs);
      uint32_t goff = (uint32_t)((gy * WW + gx) * 4);
      // GVS mode: mem = SGPR base + 32-bit VGPR offset; LDS dest addr in VGPR.
      asm volatile("global_load_async_to_lds_b32 %0, %1, %2"
                   :: "v"(lds_byte), "v"(goff), "s"(plane)
                   : "memory");
    } else {
      tile[lofs] = -__builtin_inff();
    }
  }
  asm volatile("s_wait_asynccnt 0" ::: "memory");
  __syncthreads();
}

__global__ void zero_kernel(int* __restrict__ p, int n) {
  int i = blockIdx.x * blockDim.x + threadIdx.x;
  if (i < n) p[i] = 0;
}

// Pass 1: sigmoid-order NMS keys -> per-batch 8192-bin histogram of top 13 bits.
__global__ void nms_hist_kernel(const float* __restrict__ hm,
                                int* __restrict__ hist) {
  __shared__ float tile[HY * LSTRIDE];
  const int tilesX = WW / TX;                 // 8
  const int tilesPerPlane = tilesX * (HH / TY); // 256
  int blk = blockIdx.x;
  int plane = blk / tilesPerPlane;
  int trem  = blk - plane * tilesPerPlane;
  int tyi = trem / tilesX, txi = trem - tyi * tilesX;
  int b = plane / CHN;
  int x0 = txi * TX, y0 = tyi * TY;
  const float* p = hm + (size_t)plane * HW;
  load_tile(p, x0, y0, tile);

  int tid = threadIdx.x;
  int tx  = tid & (TX - 1);
  int tyb = tid >> 6;  // 0..3, each thread covers 4 rows
  for (int i = 0; i < 4; ++i) {
    int r = tyb * 4 + i;
    const float* c = &tile[(r + 1) * LSTRIDE + (tx + 1)];
    float x = c[0];
    float m = fmaxf(fmaxf(fmaxf(c[-LSTRIDE - 1], c[-LSTRIDE]),
                          fmaxf(c[-LSTRIDE + 1], c[-1])),
                    fmaxf(fmaxf(c[1], c[LSTRIDE - 1]),
                          fmaxf(c[LSTRIDE], c[LSTRIDE + 1])));
    if (x >= m) {  // NMS keeps pixel iff it equals its 3x3 max
      uint32_t key = ord_key(x);
      atomicAdd(&hist[b * NBINS + (key >> BIN_SHIFT)], 1);
    }
  }
}

// Radix-select threshold: highest bin T such that count(bins >= T) >= K.
__global__ void thresh_kernel(const int* __restrict__ hist,
                              int* __restrict__ thresh, int K) {
  __shared__ int scan[256];
  int b = blockIdx.x;
  const int* h = hist + b * NBINS;
  int t = threadIdx.x;
  int hi = NBINS - 32 * t;   // chunk t covers bins [hi-32, hi) (top-down order)
  int s = 0;
  for (int i = 1; i <= 32; ++i) s += h[hi - i];
  scan[t] = s;
  __syncthreads();
  for (int off = 1; off < 256; off <<= 1) {
    int add = (t >= off) ? scan[t - off] : 0;
    __syncthreads();
    scan[t] += add;
    __syncthreads();
  }
  int prev = (t > 0) ? scan[t - 1] : 0;
  if (scan[t] >= K && prev < K) {
    int cum = prev;
    int tb = 0;
    for (int i = 1; i <= 32; ++i) {
      cum += h[hi - i];
      if (cum >= K) { tb = hi - i; break; }
    }
    thresh[b] = tb;
  }
  if (t == 255 && scan[255] < K) thresh[b] = 0;  // degenerate fallback
}

// Pass 2: recompute keys, append candidates in bins >= threshold.
__global__ void collect_kernel(const float* __restrict__ hm,
                               const int* __restrict__ thresh,
                               int* __restrict__ counts,
                               uint2* __restrict__ cand) {
  __shared__ float tile[HY * LSTRIDE];
  const int tilesX = WW / TX;
  const int tilesPerPlane = tilesX * (HH / TY);
  int blk = blockIdx.x;
  int plane = blk / tilesPerPlane;
  int trem  = blk - plane * tilesPerPlane;
  int tyi = trem / tilesX, txi = trem - tyi * tilesX;
  int b = plane / CHN;
  int ch = plane - b * CHN;
  int x0 = txi * TX, y0 = tyi * TY;
  const float* p = hm + (size_t)plane * HW;
  load_tile(p, x0, y0, tile);

  int T = thresh[b];
  int tid = threadIdx.x;
  int tx  = tid & (TX - 1);
  int tyb = tid >> 6;
  for (int i = 0; i < 4; ++i) {
    int r = tyb * 4 + i;
    const float* c = &tile[(r + 1) * LSTRIDE + (tx + 1)];
    float x = c[0];
    float m = fmaxf(fmaxf(fmaxf(c[-LSTRIDE - 1], c[-LSTRIDE]),
                          fmaxf(c[-LSTRIDE + 1], c[-1])),
                    fmaxf(fmaxf(c[1], c[LSTRIDE - 1]),
                          fmaxf(c[LSTRIDE], c[LSTRIDE + 1])));
    if (x >= m) {
      uint32_t key = ord_key(x);
      if ((int)(key >> BIN_SHIFT) >= T) {
        int pos = atomicAdd(&counts[b], 1);
        if (pos < CAP) {
          uint32_t sp  = (uint32_t)((y0 + r) * WW + (x0 + tx));
          uint32_t idx = (uint32_t)ch * (uint32_t)HW + sp;
          cand[b * CAP + pos] = make_uint2(key, idx);
        }
      }
    }
  }
}

// Per batch: bitonic sort candidates descending (ties -> smaller index first),
// take top K, apply sigmoid, gather features, emit (K,10) rows.
__global__ void select_gather_kernel(const uint2* __restrict__ cand,
                                     const int* __restrict__ counts,
                                     const float* __restrict__ cen,
                                     const float* __restrict__ dir,
                                     const float* __restrict__ zc,
                                     const float* __restrict__ dims,
                                     float* __restrict__ out, int K) {
  __shared__ unsigned long long s[CAP];  // 32 KB
  int b = blockIdx.x;
  int M = counts[b];
  if (M > CAP) M = CAP;
  int tid = threadIdx.x;

  for (int i = tid; i < CAP; i += blockDim.x) {
    if (i < M) {
      uint2 kv = cand[b * CAP + i];
      s[i] = ((unsigned long long)kv.x << 32) |
             (unsigned long long)(uint32_t)(~kv.y);
    } else {
      s[i] = 0ull;
    }
  }
  __syncthreads();

  for (int k = 2; k <= CAP; k <<= 1) {
    for (int j = k >> 1; j > 0; j >>= 1) {
      for (int t = tid; t < CAP; t += blockDim.x) {
        int ixj = t ^ j;
        if (ixj > t) {
          unsigned long long a = s[t], bb = s[ixj];
          bool desc = ((t & k) == 0);
          if (desc ? (a < bb) : (a > bb)) { s[t] = bb; s[ixj] = a; }
        }
      }
      __syncthreads();
    }
  }

  for (int j = tid; j < K; j += blockDim.x) {
    unsigned long long v = s[j];
    uint32_t key = (uint32_t)(v >> 32);
    uint32_t idx = ~((uint32_t)v);
    uint32_t bits = (key & 0x80000000u) ? (key ^ 0x80000000u) : ~key;
    float x = __uint_as_float(bits);
    float score = 1.0f / (1.0f + expf(-x));
    int c  = (int)(idx / (uint32_t)HW);
    int sp = (int)(idx - (uint32_t)c * (uint32_t)HW);
    float row = (float)(sp / WW);
    float col = (float)(sp % WW);

    const float* cb = cen  + (size_t)b * 2 * HW;
    const float* db = dir  + (size_t)b * 2 * HW;
    const float* zb = zc   + (size_t)b * 1 * HW;
    const float* mb = dims + (size_t)b * 3 * HW;
    float o0 = cb[sp],      o1 = cb[HW + sp];
    float d0 = db[sp],      d1 = db[HW + sp];
    float zz = zb[sp];
    float m0 = mb[sp], m1 = mb[HW + sp], m2 = mb[2 * HW + sp];

    float* o = out + ((size_t)b * K + j) * 10;
    o[0] = score;
    o[1] = col + o0;
    o[2] = row + o1;
    o[3] = zz;
    o[4] = m0; o[5] = m1; o[6] = m2;
    o[7] = d0; o[8] = d1;
    o[9] = (float)c;
  }
}

extern "C" void kernel_launch(void* const* d_in, const int* in_sizes, int n_in,
                              void* d_out, int out_size, void* d_ws, size_t ws_size,
                              hipStream_t stream) {
  (void)in_sizes; (void)n_in; (void)ws_size;
  const float* hm   = (const float*)d_in[0];
  const float* cen  = (const float*)d_in[1];
  const float* dir  = (const float*)d_in[2];
  const float* zc   = (const float*)d_in[3];
  const float* dims = (const float*)d_in[4];
  int K = out_size / (BATCH * 10);   // 500

  // Scratch layout (~1.05 MB total)
  int*   hist   = (int*)d_ws;                 // BATCH*NBINS ints
  int*   counts = hist + BATCH * NBINS;       // BATCH ints
  int*   thresh = counts + BATCH;             // BATCH ints
  uint2* cand   = (uint2*)(thresh + BATCH);   // BATCH*CAP uint2 (8B aligned)

  int nz = BATCH * NBINS + 2 * BATCH;
  zero_kernel<<<(nz + 255) / 256, 256, 0, stream>>>(hist, nz);

  int tiles = BATCH * CHN * (WW / TX) * (HH / TY);  // 12288 blocks
  nms_hist_kernel<<<tiles, 256, 0, stream>>>(hm, hist);
  thresh_kernel<<<BATCH, 256, 0, stream>>>(hist, thresh, K);
  collect_kernel<<<tiles, 256, 0, stream>>>(hm, thresh, counts, cand);
  select_gather_kernel<<<BATCH, 1024, 0, stream>>>(cand, counts, cen, dir, zc,
                                                   dims, (float*)d_out, K);
}